// LoKrLayer_70806830841898
// MI455X (gfx1250) — compile-verified
//
#include <hip/hip_runtime.h>
#include <hip/hip_bf16.h>

// LoKr fused GEMM for MI455X (gfx1250, wave32).
//
// out[m,n] = sum_k x[m,k] * ( W[n,k] + scalar*2*B[n%16]*A[n/16,k] )
//
// Roofline: W stream = 256 MB @ 23.3 TB/s ~= 11.5 us floor; fp32 WMMA
// (v_wmma_f32_16x16x4_f32) preserves reference fp32 numerics. The Kronecker
// delta is folded into the streamed WMMA B-operand, so the 256 MB delta
// matrix is never materialized and W is read from HBM exactly once.
//
// Load-width trick: WMMA is K-order-independent, so we permute the logical K
// assignment per K=8 group (half-wave h covers K = kb+4h .. kb+4h+3) and load
// float4 per lane. The SAME permutation is applied to the A-operand (x) and
// B-operand (W) loads, so the contraction is exact while every global load is
// the full-width b128 — 6 loads per 8 WMMAs instead of 12.

typedef __attribute__((ext_vector_type(2))) float v2f;
typedef __attribute__((ext_vector_type(4))) float v4f;
typedef __attribute__((ext_vector_type(8))) float v8f;

namespace {
constexpr int M_DIM = 128;
constexpr int N_DIM = 8192;
constexpr int K_DIM = 8192;
constexpr float SCALING = 2.0f;  // alpha / rank
}

// Block = 256 threads = 8 waves.
//   wave & 3   -> which of 4 consecutive N-tiles (16 cols each)
//   wave >> 2  -> which 64-row half of M (both halves share W cachelines in L0)
// Grid = 8192 / (16*4) = 128 blocks. Each wave: 4 C tiles (16x16 f32 = v8f).
__global__ __launch_bounds__(256, 1)
void lokr_fused_gemm(const float* __restrict__ x,
                     const float* __restrict__ W,
                     const float* __restrict__ A,
                     const float* __restrict__ B16,
                     const float* __restrict__ scal,
                     float* __restrict__ out)
{
    const int lane  = threadIdx.x & 31;
    const int wave  = threadIdx.x >> 5;              // 0..7
    const int nt    = blockIdx.x * 4 + (wave & 3);   // N-tile 0..511
    const int mbase = (wave >> 2) * 64;              // 0 or 64
    const int n0    = nt * 16;
    const int l16   = lane & 15;                     // N (B/C) or M (A) index
    const int lhalf = lane >> 4;                     // K-half select (0 or 1)

    // Lane-constant LoKr scale: column n = n0 + l16  ->  n % 16 == l16.
    const float sb = scal[0] * SCALING * B16[l16];

    // Per K=8 group: half-wave h loads float4 covering K = kb + 4h .. kb+4h+3.
    const float* wp = W + (size_t)(n0 + l16) * K_DIM + 4 * lhalf;     // B operand
    const float* ap = A + (size_t)nt * K_DIM + 4 * lhalf;             // delta row
    const float* xp = x + (size_t)(mbase + l16) * K_DIM + 4 * lhalf;  // A operand

    v8f c[4] = {};

    for (int kc = 0; kc < K_DIM; kc += 32) {
        // Stream-prefetch W ~8 KB ahead (speculative; drops on OOB translate).
        __builtin_prefetch(wp + kc + 2048, 0, 1);
        #pragma unroll
        for (int ku = 0; ku < 32; ku += 8) {
            const int kb = kc + ku;
            const v4f w4 = *(const v4f*)(wp + kb);
            const v4f a4 = *(const v4f*)(ap + kb);
            // W_eff^T = W + sb * A_row (delta folded in); packs to v_pk_fma_f32.
            v4f be;
            be.x = __builtin_fmaf(sb, a4.x, w4.x);
            be.y = __builtin_fmaf(sb, a4.y, w4.y);
            be.z = __builtin_fmaf(sb, a4.z, w4.z);
            be.w = __builtin_fmaf(sb, a4.w, w4.w);
            const v2f b0 = __builtin_shufflevector(be, be, 0, 1);
            const v2f b1 = __builtin_shufflevector(be, be, 2, 3);

            v4f x4[4];
            #pragma unroll
            for (int mt = 0; mt < 4; ++mt)
                x4[mt] = *(const v4f*)(xp + (size_t)(mt * 16) * K_DIM + kb);

            #pragma unroll
            for (int mt = 0; mt < 4; ++mt) {
                const v2f a0 = __builtin_shufflevector(x4[mt], x4[mt], 0, 1);
                c[mt] = __builtin_amdgcn_wmma_f32_16x16x4_f32(
                    false, a0, false, b0, (short)0, c[mt], false, false);
            }
            #pragma unroll
            for (int mt = 0; mt < 4; ++mt) {
                const v2f a1 = __builtin_shufflevector(x4[mt], x4[mt], 2, 3);
                c[mt] = __builtin_amdgcn_wmma_f32_16x16x4_f32(
                    false, a1, false, b1, (short)0, c[mt], false, false);
            }
        }
    }

    // C/D layout: VGPR r -> M = r (lanes 0-15) / M = 8+r (lanes 16-31); N = l16.
    float* op = out + n0 + l16;
    #pragma unroll
    for (int mt = 0; mt < 4; ++mt) {
        #pragma unroll
        for (int r = 0; r < 8; ++r) {
            const int m = mbase + mt * 16 + lhalf * 8 + r;
            op[(size_t)m * N_DIM] = c[mt][r];
        }
    }
}

extern "C" void kernel_launch(void* const* d_in, const int* in_sizes, int n_in,
                              void* d_out, int out_size, void* d_ws, size_t ws_size,
                              hipStream_t stream) {
    (void)in_sizes; (void)n_in; (void)out_size; (void)d_ws; (void)ws_size;
    const float* x    = (const float*)d_in[0];   // (128, 8192)
    const float* W    = (const float*)d_in[1];   // (8192, 8192)
    const float* A    = (const float*)d_in[2];   // (512, 8192)
    const float* B16  = (const float*)d_in[3];   // (16, 1)
    const float* scal = (const float*)d_in[4];   // (1,)
    float* out = (float*)d_out;                  // (128, 8192)

    dim3 grid(N_DIM / 64);   // 128 blocks: 4 N-tiles * 2 M-halves per block
    dim3 block(256);         // 8 waves (wave32)
    hipLaunchKernelGGL(lokr_fused_gemm, grid, block, 0, stream,
                       x, W, A, B16, scal, out);
}